// DC_CE_Marginal_loss_154618822767
// MI455X (gfx1250) — compile-verified
//
#include <hip/hip_runtime.h>
#include <math.h>

// Problem constants (B,C,D,H,W) = (2,8,64,160,160)
#define Bn 2
#define Cn 8
#define PLANE  1638400   // D*H*W
#define PLANE4 409600    // PLANE/4
#define NBLK 512         // blocks per sample (label + main passes)
#define TPB 256

typedef __attribute__((ext_vector_type(2))) float v2f;
typedef __attribute__((ext_vector_type(8))) float v8f;

// ---------------- kernel 0: zero the integer count scratch ----------------
__global__ void zero_counts_kernel(int* counts) {
    if (threadIdx.x < Bn * Cn) counts[threadIdx.x] = 0;
}

// ---------------- kernel 1: label-compression pass ------------------------
// Reads the one-hot target once (105 MB), writes 1 byte/voxel labels
// (3.2 MB) and exact integer per-class counts (== ref_vol). Integer
// atomics are order-independent -> deterministic.
__global__ __launch_bounds__(TPB) void label_kernel(const float* __restrict__ target,
                                                    unsigned int* __restrict__ labels32,
                                                    int* __restrict__ counts) {
    const int b = blockIdx.y;
    const float* tb = target + (size_t)b * Cn * PLANE;
    unsigned int* lb = labels32 + (size_t)b * PLANE4;

    int cnt[Cn];
#pragma unroll
    for (int c = 0; c < Cn; c++) cnt[c] = 0;

    const int stride = gridDim.x * TPB;
    for (int i = blockIdx.x * TPB + threadIdx.x; i < PLANE4; i += stride) {
        float4 tv[Cn];
#pragma unroll
        for (int c = 0; c < Cn; c++) {
            const float4* tp = reinterpret_cast<const float4*>(tb + (size_t)c * PLANE);
            tv[c] = tp[i];
            __builtin_prefetch(tp + i + stride, 0, 1);
        }
        unsigned int pack = 0;
#pragma unroll
        for (int j = 0; j < 4; j++) {
            unsigned int lab = 0;
#pragma unroll
            for (int c = 0; c < Cn; c++) {
                int hit = ((&tv[c].x)[j] != 0.f);
                lab += (unsigned int)c * (unsigned int)hit;  // strict one-hot
                cnt[c] += hit;
            }
            pack |= lab << (8 * j);
        }
        lb[i] = pack;
    }
    // per-wave integer reduce, lane 0 of each wave commits with atomicAdd
#pragma unroll
    for (int c = 0; c < Cn; c++) {
        int v = cnt[c];
        for (int off = 16; off; off >>= 1) v += __shfl_xor(v, off, 32);
        if ((threadIdx.x & 31) == 0) atomicAdd(&counts[b * Cn + c], v);
    }
}

// ---------------- kernel 2: presence mask, padding stats (tiny) ----------
__global__ void prep_kernel(const int* __restrict__ counts, int* maskArr,
                            int* padFlag, float* logpad) {
    if (threadIdx.x == 0) {
        int nb[Bn];
        int L = 0;
        for (int b = 0; b < Bn; b++) {
            int mask = 0, n = 0;
            for (int c = 0; c < Cn; c++)
                if (counts[b * Cn + c] > 0) { mask |= (1 << c); n++; }
            maskArr[b] = mask; nb[b] = n; if (n > L) L = n;
        }
        for (int b = 0; b < Bn; b++) {
            int pad = L - nb[b];
            padFlag[b] = (pad > 0) ? 1 : 0;
            logpad[b] = logf((float)(pad > 0 ? pad : 1));
        }
    }
}

// ---------------- kernel 3: main streaming pass (net + compact labels) ----
__global__ __launch_bounds__(TPB) void main_kernel(
    const float* __restrict__ net, const unsigned int* __restrict__ labels32,
    const int* __restrict__ maskArr, const int* __restrict__ padFlag,
    const float* __restrict__ logpadArr,
    float* __restrict__ ce_part, float* __restrict__ seg_part,
    float* __restrict__ int_part) {
    const int b = blockIdx.y;
    const int pmask = maskArr[b];
    const int pf = padFlag[b];
    const float logpad = logpadArr[b];
    const float* nb = net + (size_t)b * Cn * PLANE;
    const unsigned int* lb = labels32 + (size_t)b * PLANE4;

    float ce_acc = 0.f;
    float seg_acc[Cn], int_acc[Cn];
#pragma unroll
    for (int c = 0; c < Cn; c++) { seg_acc[c] = 0.f; int_acc[c] = 0.f; }

    const int stride = gridDim.x * TPB;
    for (int i = blockIdx.x * TPB + threadIdx.x; i < PLANE4; i += stride) {
        float4 xv[Cn];
#pragma unroll
        for (int c = 0; c < Cn; c++) {
            const float4* xp = reinterpret_cast<const float4*>(nb + (size_t)c * PLANE);
            xv[c] = xp[i];
            __builtin_prefetch(xp + i + stride, 0, 1);   // global_prefetch_b8
        }
        unsigned int labv = lb[i];                        // 4 voxels' labels
        __builtin_prefetch(lb + i + stride, 0, 1);
#pragma unroll
        for (int j = 0; j < 4; j++) {
            const unsigned int lab = (labv >> (8 * j)) & 0xFFu;
            float x[Cn];
#pragma unroll
            for (int c = 0; c < Cn; c++) x[c] = (&xv[c].x)[j];
            // merge absent foreground logits into background
            float bg = 0.f;
#pragma unroll
            for (int c = 1; c < Cn; c++)
                if (!((pmask >> c) & 1)) bg += x[c];
            float m[Cn];
            m[0] = x[0] + bg;
#pragma unroll
            for (int c = 1; c < Cn; c++) m[c] = x[c];
            // masked max
            float mx = -3.4e38f;
#pragma unroll
            for (int c = 0; c < Cn; c++)
                if ((pmask >> c) & 1) mx = fmaxf(mx, m[c]);
            // masked softmax / LSE
            float se = 0.f, pv[Cn];
#pragma unroll
            for (int c = 0; c < Cn; c++) {
                if ((pmask >> c) & 1) { float e = __expf(m[c] - mx); pv[c] = e; se += e; }
                else pv[c] = 0.f;
            }
            float lse = mx + __logf(se);
            if (pf) { // logaddexp with log(pad) for zero-logit padding channels
                float M = fmaxf(lse, logpad);
                lse = M + __logf(__expf(lse - M) + __expf(logpad - M));
            }
            // sum(t_eff * m) == m[label] (label class is present by construction)
            float tm = 0.f;
#pragma unroll
            for (int c = 0; c < Cn; c++) tm += (lab == (unsigned int)c) ? m[c] : 0.f;
            ce_acc += lse - tm;
            float inv = 1.0f / se;
#pragma unroll
            for (int c = 0; c < Cn; c++) {
                float p = pv[c] * inv;                   // absent: pv==0 -> p==0
                seg_acc[c] += p;
                int_acc[c] += (lab == (unsigned int)c) ? p : 0.f;
            }
        }
    }

    // deterministic block reduction -> fixed per-block slots (no float atomics)
    __shared__ float smem[TPB / 32];
    auto blockReduce = [&](float v) -> float {
        for (int off = 16; off; off >>= 1) v += __shfl_xor(v, off, 32);
        __syncthreads();
        if ((threadIdx.x & 31) == 0) smem[threadIdx.x >> 5] = v;
        __syncthreads();
        float r = 0.f;
        if (threadIdx.x == 0)
            for (int w = 0; w < TPB / 32; w++) r += smem[w];
        return r;
    };
    float r = blockReduce(ce_acc);
    if (threadIdx.x == 0) ce_part[b * NBLK + blockIdx.x] = r;
#pragma unroll
    for (int c = 0; c < Cn; c++) {
        r = blockReduce(seg_acc[c]);
        if (threadIdx.x == 0) seg_part[(b * Cn + c) * NBLK + blockIdx.x] = r;
        r = blockReduce(int_acc[c]);
        if (threadIdx.x == 0) int_part[(b * Cn + c) * NBLK + blockIdx.x] = r;
    }
}

// ---------------- kernel 4: final fold via fp32 WMMA ----------------------
// Sum 512 partials with one wave32: 16 serial adds/lane, then one
// V_WMMA_F32_16X16X4_F32 with A = ones folds the 32 lane partials into
// exact fp32 column sums, finished with a 16-lane xor-shuffle tree.
__device__ inline float waveSum512(const float* __restrict__ base) {
    const int lane = threadIdx.x & 31;
    float s = 0.f;
#pragma unroll
    for (int k = 0; k < 16; k++) s += base[lane * 16 + k];
    v2f a; a[0] = 1.f; a[1] = 1.f;     // A(16x4) = all ones
    v2f bv; bv[0] = s; bv[1] = 0.f;    // B rows 0..1 carry the 32 lane partials
    v8f cz = {};
    v8f d = __builtin_amdgcn_wmma_f32_16x16x4_f32(
        false, a, false, bv, (short)0, cz, false, false);
    float v = d[0];                    // D row: fp32 column sums of B
    v += __shfl_xor(v, 1, 32);
    v += __shfl_xor(v, 2, 32);
    v += __shfl_xor(v, 4, 32);
    v += __shfl_xor(v, 8, 32);
    return v;                          // uniform across the wave
}

__global__ void final_kernel(const int* __restrict__ counts,
                             const float* __restrict__ ce_part,
                             const float* __restrict__ seg_part,
                             const float* __restrict__ int_part,
                             float* __restrict__ out) {
    float ce_sum = 0.f;
    for (int b = 0; b < Bn; b++) ce_sum += waveSum512(ce_part + b * NBLK);
    float ce_loss = ce_sum / (float)((long long)Bn * PLANE);

    float dc_sum = 0.f;
    for (int b = 0; b < Bn; b++) {
        float dsum = 0.f, nbv = 0.f;
        for (int c = 0; c < Cn; c++) {
            int cnt = counts[b * Cn + c];   // uniform -> EXEC stays all-1 for WMMA
            if (cnt > 0) {
                float seg  = waveSum512(seg_part + (b * Cn + c) * NBLK);
                float intr = waveSum512(int_part + (b * Cn + c) * NBLK);
                float dice = 2.f * intr / ((float)cnt + seg + 1e-5f);
                dsum += dice; nbv += 1.f;
            }
        }
        dc_sum += 1.f - dsum / nbv;
    }
    if (threadIdx.x == 0) out[0] = 0.5f * ce_loss + 0.5f * dc_sum / (float)Bn;
}

// ---------------- host launch --------------------------------------------
extern "C" void kernel_launch(void* const* d_in, const int* in_sizes, int n_in,
                              void* d_out, int out_size, void* d_ws, size_t ws_size,
                              hipStream_t stream) {
    const float* net    = (const float*)d_in[0];
    const float* target = (const float*)d_in[1];
    char* ws = (char*)d_ws;
    // workspace layout (16B-aligned; ~70 KB partials + 3.2 MB labels)
    int*   counts  = (int*)(ws + 0);                       // 16 ints
    int*   maskArr = (int*)(ws + 64);                      // 2 ints
    int*   padFlag = (int*)(ws + 96);                      // 2 ints
    float* logpad  = (float*)(ws + 128);                   // 2 floats
    float* ce_part  = (float*)(ws + 256);                                // B*NBLK
    float* seg_part = (float*)(ws + 256 + 4 * Bn * NBLK);                // B*C*NBLK
    float* int_part = (float*)(ws + 256 + 4 * Bn * NBLK + 4 * Bn * Cn * NBLK);
    unsigned int* labels32 = (unsigned int*)(ws + 70656);  // Bn*PLANE bytes (packed u8 labels)
    float* out = (float*)d_out;

    zero_counts_kernel<<<1, 32, 0, stream>>>(counts);
    label_kernel<<<dim3(NBLK, Bn), TPB, 0, stream>>>(target, labels32, counts);
    prep_kernel<<<1, 32, 0, stream>>>(counts, maskArr, padFlag, logpad);
    main_kernel<<<dim3(NBLK, Bn), TPB, 0, stream>>>(net, labels32, maskArr, padFlag,
                                                    logpad, ce_part, seg_part, int_part);
    final_kernel<<<1, 32, 0, stream>>>(counts, ce_part, seg_part, int_part, out);
}